// Model_65352222376400
// MI455X (gfx1250) — compile-verified
//
#include <hip/hip_runtime.h>
#include <cstdint>
#include <cstddef>

typedef __attribute__((ext_vector_type(2))) float v2f;
typedef __attribute__((ext_vector_type(8))) float v8f;

#define NN_NODES 9000
#define DIS_N    3000
#define SNO_N    6000
#define NEDGE    288000
#define F1_DIM   512
#define F2_DIM   256
#define F3_DIM   128

#define BM 128
#define BN 64
#define BK 16
// LDS row strides (in float2 units): ≡16 (mod 32) so that the two lane-halves
// (which read adjacent k-pair rows) hit disjoint 32-bank halves.
#define ASTRIDE 144
#define BSTRIDE 80

static __device__ inline v8f wmma4(v2f a, v2f b, v8f c) {
  // D = A(16x4,f32) * B(4x16,f32) + C(16x16,f32)
  return __builtin_amdgcn_wmma_f32_16x16x4_f32(false, a, false, b, (short)0, c,
                                               false, false);
}

// ---------------------------------------------------------------------------
// fp32 WMMA GEMM: C[M,N] = A[M,K] @ B  (B is [K,N] or, transB, [N,K])
// Block tile 128x64, 256 threads = 8 waves; wave w -> rows w*16..+16, all 64
// cols (4 accum frags sharing one A frag). LDS holds K-paired float2 tiles so
// each WMMA operand is one ds_load_b64.
// ---------------------------------------------------------------------------
__global__ __launch_bounds__(256) void gemm_wmma_f32(
    const float* __restrict__ A, const float* __restrict__ B,
    float* __restrict__ C, int M, int N, int K,
    int lda, int ldb, int ldc, int transB) {
  __shared__ v2f As2[BK / 2][ASTRIDE];  // [k-pair][row]
  __shared__ v2f Bs2[BK / 2][BSTRIDE];  // [k-pair][col]

  const int tid  = threadIdx.x;
  const int lane = tid & 31;
  const int wave = tid >> 5;                 // 0..7 : 16-row band
  const int blockRow = blockIdx.y * BM;
  const int blockCol = blockIdx.x * BN;
  const int ml = lane & 15;
  const int kq = (lane >> 4) << 1;           // K sub-offset 0 or 2 (ISA layout)

  v8f acc[4];
#pragma unroll
  for (int c = 0; c < 4; ++c) acc[c] = (v8f){0.f,0.f,0.f,0.f,0.f,0.f,0.f,0.f};

  const int i0 = tid >> 2;                   // 0..63 (A row / B col group)
  const int kc = (tid & 3) << 2;             // k-chunk 0,4,8,12
  const int bkk = tid >> 4;                  // 0..15 (B fast path k row)
  const int bj4 = (tid & 15) << 2;           // 0..60 (B fast path col*4)

  const bool fullMN = (blockRow + BM <= M) && (blockCol + BN <= N) &&
                      ((lda & 3) == 0) && ((ldb & 3) == 0);

  for (int k0 = 0; k0 < K; k0 += BK) {
    __syncthreads();
    if (fullMN && (k0 + BK <= K)) {
      // -------- fast path: unguarded b128 loads --------
#pragma unroll
      for (int r = 0; r < BM / 64; ++r) {
        int i = i0 + (r << 6);
        const float* gp = A + (size_t)(blockRow + i) * lda + k0 + kc;
        float4 v = *(const float4*)gp;
        v2f t0 = {v.x, v.y}, t1 = {v.z, v.w};
        As2[(kc >> 1)][i]     = t0;
        As2[(kc >> 1) + 1][i] = t1;
      }
      if (transB) {
        const float* gp = B + (size_t)(blockCol + i0) * ldb + k0 + kc;
        float4 v = *(const float4*)gp;
        v2f t0 = {v.x, v.y}, t1 = {v.z, v.w};
        Bs2[(kc >> 1)][i0]     = t0;
        Bs2[(kc >> 1) + 1][i0] = t1;
      } else {
        const float* gp = B + (size_t)(k0 + bkk) * ldb + blockCol + bj4;
        float4 v = *(const float4*)gp;
        float* p0 = (float*)&Bs2[bkk >> 1][bj4];
        int par = bkk & 1;
        p0[0 + par] = v.x; p0[2 + par] = v.y;
        p0[4 + par] = v.z; p0[6 + par] = v.w;
      }
      if (k0 + BK < K) {  // prefetch next K-tile (global_prefetch_b8)
        __builtin_prefetch(A + (size_t)(blockRow + i0) * lda + (k0 + BK) + kc, 0, 3);
        __builtin_prefetch(transB
                               ? B + (size_t)(blockCol + i0) * ldb + (k0 + BK) + kc
                               : B + (size_t)(k0 + BK + bkk) * ldb + blockCol + bj4,
                           0, 3);
      }
    } else {
      // -------- slow path: guarded scalar loads (edge tiles / K tail) ------
#pragma unroll
      for (int r = 0; r < BM / 64; ++r) {
        int i = i0 + (r << 6);
        int grow = blockRow + i;
        float e[4];
#pragma unroll
        for (int s = 0; s < 4; ++s) {
          int k = k0 + kc + s;
          e[s] = (grow < M && k < K) ? A[(size_t)grow * lda + k] : 0.f;
        }
        v2f t0 = {e[0], e[1]}, t1 = {e[2], e[3]};
        As2[(kc >> 1)][i]     = t0;
        As2[(kc >> 1) + 1][i] = t1;
      }
      {
        int gcol = blockCol + i0;
        float e[4];
#pragma unroll
        for (int s = 0; s < 4; ++s) {
          int k = k0 + kc + s;
          e[s] = (gcol < N && k < K)
                     ? (transB ? B[(size_t)gcol * ldb + k]
                               : B[(size_t)k * ldb + gcol])
                     : 0.f;
        }
        v2f t0 = {e[0], e[1]}, t1 = {e[2], e[3]};
        Bs2[(kc >> 1)][i0]     = t0;
        Bs2[(kc >> 1) + 1][i0] = t1;
      }
    }
    __syncthreads();
#pragma unroll
    for (int kk = 0; kk < BK; kk += 4) {
      int kp = (kk + kq) >> 1;
      v2f a  = As2[kp][(wave << 4) + ml];
      v2f b0 = Bs2[kp][ml];
      v2f b1 = Bs2[kp][16 + ml];
      v2f b2 = Bs2[kp][32 + ml];
      v2f b3 = Bs2[kp][48 + ml];
      acc[0] = wmma4(a, b0, acc[0]);
      acc[1] = wmma4(a, b1, acc[1]);
      acc[2] = wmma4(a, b2, acc[2]);
      acc[3] = wmma4(a, b3, acc[3]);
    }
  }

  // C/D layout: VGPR v -> row v + 8*(lane>=16); col = lane&15
  const int r0 = blockRow + (wave << 4) + ((lane >> 4) << 3);
  const int cc = blockCol + ml;
#pragma unroll
  for (int v = 0; v < 8; ++v) {
    int r = r0 + v;
    if (r < M) {
#pragma unroll
      for (int c = 0; c < 4; ++c) {
        int col = cc + (c << 4);
        if (col < N) C[(size_t)r * ldc + col] = acc[c][v];
      }
    }
  }
}

// ---------------------------------------------------------------------------
// Elementwise / graph kernels
// ---------------------------------------------------------------------------
__global__ void fill_f32(float* p, long long n, float v) {
  long long i = blockIdx.x * (long long)blockDim.x + threadIdx.x;
  long long st = (long long)gridDim.x * blockDim.x;
  for (; i < n; i += st) p[i] = v;
}

__global__ void scale_copy(float* dst, const float* __restrict__ src,
                           long long n, float s) {
  long long i = blockIdx.x * (long long)blockDim.x + threadIdx.x;
  long long st = (long long)gridDim.x * blockDim.x;
  for (; i < n; i += st) dst[i] = s * src[i];
}

__global__ void gather_w(const float* __restrict__ adj, const int* __restrict__ src,
                         const int* __restrict__ dst, float* w, int e, int n) {
  int i = blockIdx.x * blockDim.x + threadIdx.x;
  if (i < e) w[i] = adj[(size_t)src[i] * n + dst[i]];
}

__global__ void deg_acc(const float* __restrict__ w, const int* __restrict__ dst,
                        float* deg, int e) {
  int i = blockIdx.x * blockDim.x + threadIdx.x;
  if (i < e) atomicAdd(&deg[dst[i]], w[i]);
}

__global__ void dinv_k(const float* __restrict__ deg, float* dinv, int n) {
  int i = blockIdx.x * blockDim.x + threadIdx.x;
  if (i < n) { float d = deg[i]; dinv[i] = (d > 0.f) ? rsqrtf(d) : 0.f; }
}

__global__ void edge_norm(const float* __restrict__ w, const int* __restrict__ src,
                          const int* __restrict__ dst, const float* __restrict__ dinv,
                          float* nrm, int e) {
  int i = blockIdx.x * blockDim.x + threadIdx.x;
  if (i < e) nrm[i] = dinv[src[i]] * w[i] * dinv[dst[i]];
}

// 4 features per thread: one float4 h read + 4 f32 atomics (L2-resident acc).
__global__ void edge_scatter4(const float* __restrict__ h, const float* __restrict__ nrm,
                              const int* __restrict__ src, const int* __restrict__ dst,
                              float* acc, int F, long long total4) {
  long long idx = blockIdx.x * (long long)blockDim.x + threadIdx.x;
  if (idx >= total4) return;
  int F4 = F >> 2;
  int e  = (int)(idx / F4);
  int f  = (int)(idx - (long long)e * F4) << 2;
  float nv = nrm[e];
  const float4 hv = *(const float4*)(h + (size_t)src[e] * F + f);
  float* ap = acc + (size_t)dst[e] * F + f;
  atomicAdd(ap + 0, hv.x * nv);
  atomicAdd(ap + 1, hv.y * nv);
  atomicAdd(ap + 2, hv.z * nv);
  atomicAdd(ap + 3, hv.w * nv);
}

__global__ void self_loop(const float* __restrict__ h, const float* __restrict__ dinv,
                          float* acc, int F, long long n) {
  long long idx = blockIdx.x * (long long)blockDim.x + threadIdx.x;
  if (idx >= n) return;
  int i = (int)(idx / F);
  float dv = dinv[i];
  acc[idx] += h[idx] * dv * dv;
}

__global__ void bias_relu(const float* __restrict__ acc, const float* __restrict__ b,
                          float* H, int F, long long n) {
  long long idx = blockIdx.x * (long long)blockDim.x + threadIdx.x;
  if (idx >= n) return;
  int f = (int)(idx % F);
  float v = acc[idx] + b[f];
  H[idx] = v > 0.f ? v : 0.f;
}

__global__ __launch_bounds__(256) void rowminmax(const float* __restrict__ H,
                                                 float* __restrict__ Y, int F) {
  int row = blockIdx.x;
  const float* hr = H + (size_t)row * F;
  float mn = 3.402823466e38f, mx = -3.402823466e38f;
  for (int f = threadIdx.x; f < F; f += 256) {
    float v = hr[f];
    mn = fminf(mn, v);
    mx = fmaxf(mx, v);
  }
  __shared__ float smn[256], smx[256];
  smn[threadIdx.x] = mn;
  smx[threadIdx.x] = mx;
  __syncthreads();
  for (int s = 128; s > 0; s >>= 1) {
    if (threadIdx.x < s) {
      smn[threadIdx.x] = fminf(smn[threadIdx.x], smn[threadIdx.x + s]);
      smx[threadIdx.x] = fmaxf(smx[threadIdx.x], smx[threadIdx.x + s]);
    }
    __syncthreads();
  }
  float lo = smn[0];
  float rng = smx[0] - lo;
  float den = (rng == 0.f) ? 1.f : rng;
  float* yr = Y + (size_t)row * F;
  for (int f = threadIdx.x; f < F; f += 256) yr[f] = (hr[f] - lo) / den;
}

__global__ void init_scalars(float* ssum, unsigned int* minb) {
  if (threadIdx.x == 0 && blockIdx.x == 0) {
    *ssum = 0.f;
    *minb = 0x7F800000u;  // +inf
  }
}

__global__ void diag_sum(const float* __restrict__ K, int n, float* ssum) {
  int i = blockIdx.x * blockDim.x + threadIdx.x;
  if (i < n) atomicAdd(ssum, K[(size_t)i * n + i]);
}

__global__ void gip_accum(const float* __restrict__ K, int n, float gamma,
                          const float* __restrict__ ssum, float* Nacc, long long nn) {
  long long idx = blockIdx.x * (long long)blockDim.x + threadIdx.x;
  if (idx >= nn) return;
  int i = (int)(idx / n);
  int j = (int)(idx % n);
  float md = *ssum / (float)n;
  float kij = K[idx] / md;
  float kii = K[(size_t)i * n + i] / md;
  float kjj = K[(size_t)j * n + j] / md;
  float dist = kii + kjj - 2.f * kij;
  Nacc[idx] += 0.25f * expf(-gamma * dist);
}

__global__ __launch_bounds__(256) void abs_minnz(float* K, long long n,
                                                 unsigned int* minb) {
  long long i = blockIdx.x * (long long)blockDim.x + threadIdx.x;
  unsigned int local = 0x7F800000u;
  if (i < n) {
    float v = fabsf(K[i]);
    K[i] = v;
    if (v > 0.f) local = __float_as_uint(v);  // positive floats: bits ordered
  }
  __shared__ unsigned int sm[256];
  sm[threadIdx.x] = local;
  __syncthreads();
  for (int s = 128; s > 0; s >>= 1) {
    if (threadIdx.x < s) {
      unsigned int a = sm[threadIdx.x], b = sm[threadIdx.x + s];
      sm[threadIdx.x] = a < b ? a : b;
    }
    __syncthreads();
  }
  if (threadIdx.x == 0) atomicMin(minb, sm[0]);
}

__global__ void diag_extract(const float* __restrict__ K, int n,
                             const unsigned int* __restrict__ minb, float* diag) {
  int i = blockIdx.x * blockDim.x + threadIdx.x;
  if (i >= n) return;
  float v = K[(size_t)i * n + i];
  if (v == 0.f) v = __uint_as_float(*minb);
  diag[i] = v;
}

__global__ void col_norm(float* K, int n, const unsigned int* __restrict__ minb,
                         const float* __restrict__ diag, long long nn) {
  long long idx = blockIdx.x * (long long)blockDim.x + threadIdx.x;
  if (idx >= nn) return;
  int j = (int)(idx % n);
  float v = K[idx];
  if (v == 0.f) v = __uint_as_float(*minb);
  K[idx] = v / diag[j];
}

__global__ void combine(const float* __restrict__ C1, const float* __restrict__ C2,
                        float* out, long long total) {
  long long idx = blockIdx.x * (long long)blockDim.x + threadIdx.x;
  if (idx >= total) return;
  int i = (int)(idx / SNO_N);
  int j = (int)(idx % SNO_N);
  out[idx] = 0.5f * (C1[idx] + C2[(size_t)j * DIS_N + i]);
}

// ---------------------------------------------------------------------------
extern "C" void kernel_launch(void* const* d_in, const int* in_sizes, int n_in,
                              void* d_out, int out_size, void* d_ws, size_t ws_size,
                              hipStream_t stream) {
  (void)in_sizes; (void)n_in; (void)out_size; (void)ws_size;
  const float* x       = (const float*)d_in[0];
  const float* adj     = (const float*)d_in[1];
  const int*   ei      = (const int*)d_in[2];
  const float* W1      = (const float*)d_in[3];
  const float* b1      = (const float*)d_in[4];
  const float* W2      = (const float*)d_in[5];
  const float* b2      = (const float*)d_in[6];
  const float* W3      = (const float*)d_in[7];
  const float* b3      = (const float*)d_in[8];
  const float* dis_sim = (const float*)d_in[9];
  const float* sno_sim = (const float*)d_in[10];
  const float* alpha1  = (const float*)d_in[11];
  const float* alpha2  = (const float*)d_in[12];
  float* out = (float*)d_out;
  const int* src = ei;
  const int* dst = ei + NEDGE;

  // Workspace carve-out (~535 MB total).
  size_t off = 0;
  auto alloc = [&](size_t nfloats) -> float* {
    float* p = (float*)((char*)d_ws + off);
    off += nfloats * sizeof(float);
    off = (off + 255) & ~(size_t)255;
    return p;
  };
  float* w    = alloc(NEDGE);
  float* deg  = alloc(NN_NODES);
  float* dinv = alloc(NN_NODES);
  float* nrm  = alloc(NEDGE);
  float* acc  = alloc((size_t)NN_NODES * F1_DIM);
  float* H1   = alloc((size_t)NN_NODES * F1_DIM);
  float* H2   = alloc((size_t)NN_NODES * F2_DIM);
  float* H3   = alloc((size_t)NN_NODES * F3_DIM);
  float* ynrm = alloc((size_t)SNO_N * F1_DIM);
  float* Kbuf = alloc((size_t)SNO_N * SNO_N);
  float* Nd   = alloc((size_t)DIS_N * DIS_N);
  float* Ns   = alloc((size_t)SNO_N * SNO_N);
  float* C1   = alloc((size_t)DIS_N * SNO_N);
  float* C2   = alloc((size_t)SNO_N * DIS_N);
  float* diag = alloc(SNO_N);
  float* ssum = alloc(1);
  unsigned int* minb = (unsigned int*)alloc(1);
  float* hbuf = C1;  // h = X@W scratch; C1 is free until the final GEMMs

  auto cdiv = [](long long a, long long b) { return (int)((a + b - 1) / b); };
  auto gemm = [&](const float* A, const float* B, float* C, int M, int Nc, int K,
                  int lda, int ldb, int ldc, int tB) {
    dim3 g((unsigned)((Nc + BN - 1) / BN), (unsigned)((M + BM - 1) / BM));
    gemm_wmma_f32<<<g, 256, 0, stream>>>(A, B, C, M, Nc, K, lda, ldb, ldc, tB);
  };

  // ---- edge preprocessing (shared by all 3 GCN layers) ----
  gather_w<<<cdiv(NEDGE, 256), 256, 0, stream>>>(adj, src, dst, w, NEDGE, NN_NODES);
  fill_f32<<<1024, 256, 0, stream>>>(deg, NN_NODES, 1.0f);  // self-loop weight 1
  deg_acc<<<cdiv(NEDGE, 256), 256, 0, stream>>>(w, dst, deg, NEDGE);
  dinv_k<<<cdiv(NN_NODES, 256), 256, 0, stream>>>(deg, dinv, NN_NODES);
  edge_norm<<<cdiv(NEDGE, 256), 256, 0, stream>>>(w, src, dst, dinv, nrm, NEDGE);

  // ---- GCN layers ----
  auto gcn = [&](const float* Xin, int Fin, const float* W, const float* b,
                 float* Hout, int Fout) {
    gemm(Xin, W, hbuf, NN_NODES, Fout, Fin, Fin, Fout, Fout, 0);
    long long nf = (long long)NN_NODES * Fout;
    fill_f32<<<4096, 256, 0, stream>>>(acc, nf, 0.f);
    long long tot4 = (long long)NEDGE * (Fout >> 2);
    edge_scatter4<<<cdiv(tot4, 256), 256, 0, stream>>>(hbuf, nrm, src, dst, acc,
                                                       Fout, tot4);
    self_loop<<<cdiv(nf, 256), 256, 0, stream>>>(hbuf, dinv, acc, Fout, nf);
    bias_relu<<<cdiv(nf, 256), 256, 0, stream>>>(acc, b, Hout, Fout, nf);
  };
  gcn(x, NN_NODES, W1, b1, H1, F1_DIM);
  gcn(H1, F1_DIM, W2, b2, H2, F2_DIM);
  gcn(H2, F2_DIM, W3, b3, H3, F3_DIM);

  // ---- GIP kernels accumulated into N_d / N_s ----
  auto gip = [&](const float* H, int rowOff, int n, int F, float gamma,
                 float* Nacc) {
    const float* Hp = H + (size_t)rowOff * F;
    rowminmax<<<n, 256, 0, stream>>>(Hp, ynrm, F);
    gemm(ynrm, ynrm, Kbuf, n, n, F, F, F, n, /*transB=*/1);  // K = y @ y^T
    init_scalars<<<1, 1, 0, stream>>>(ssum, minb);
    diag_sum<<<cdiv(n, 256), 256, 0, stream>>>(Kbuf, n, ssum);
    long long nn = (long long)n * n;
    gip_accum<<<cdiv(nn, 256), 256, 0, stream>>>(Kbuf, n, gamma, ssum, Nacc, nn);
  };
  scale_copy<<<4096, 256, 0, stream>>>(Nd, dis_sim, (long long)DIS_N * DIS_N, 0.25f);
  gip(H1, 0, DIS_N, F1_DIM, 0.5f, Nd);
  gip(H2, 0, DIS_N, F2_DIM, 0.5f, Nd);
  gip(H3, 0, DIS_N, F3_DIM, 0.5f, Nd);
  scale_copy<<<4096, 256, 0, stream>>>(Ns, sno_sim, (long long)SNO_N * SNO_N, 0.25f);
  gip(H1, DIS_N, SNO_N, F1_DIM, 0.5f, Ns);
  gip(H2, DIS_N, SNO_N, F2_DIM, 0.5f, Ns);
  gip(H3, DIS_N, SNO_N, F3_DIM, 0.5f, Ns);

  // ---- normalized_kernel (in place) ----
  auto normk = [&](float* Na, int n) {
    long long nn = (long long)n * n;
    init_scalars<<<1, 1, 0, stream>>>(ssum, minb);
    abs_minnz<<<cdiv(nn, 256), 256, 0, stream>>>(Na, nn, minb);
    diag_extract<<<cdiv(n, 256), 256, 0, stream>>>(Na, n, minb, diag);
    col_norm<<<cdiv(nn, 256), 256, 0, stream>>>(Na, n, minb, diag, nn);
  };
  normk(Nd, DIS_N);
  normk(Ns, SNO_N);

  // ---- final projection ----
  gemm(Nd, alpha1, C1, DIS_N, SNO_N, DIS_N, DIS_N, SNO_N, SNO_N, 0);
  gemm(Ns, alpha2, C2, SNO_N, DIS_N, SNO_N, SNO_N, DIS_N, DIS_N, 0);
  long long tot = (long long)DIS_N * SNO_N;
  combine<<<cdiv(tot, 256), 256, 0, stream>>>(C1, C2, out, tot);
}